// MACG_34815004902096
// MI455X (gfx1250) — compile-verified
//
#include <hip/hip_runtime.h>
#include <cmath>

typedef __attribute__((ext_vector_type(2))) float v2f;
typedef __attribute__((ext_vector_type(4))) float v4f;
typedef __attribute__((ext_vector_type(8))) float v8f;

#define NP 100        // P
#define NQ 2          // Q
#define NR 25         // R
#define NK 10         // K components
#define COLS 256      // K*R = 250, padded to 256 (16 tiles of 16)
#define KSTEPS 25     // P / 4 (WMMA K=4 steps)
#define SPW 16        // samples per wave (two 16-row A tiles)

// Workspace layout (in floats)
#define WS_CONST0 0                  // logSA_Stiefel
#define WS_LD 16                     // log det Sigma_k, K entries
#define WS_LP 32                     // log_softmax(pi), K entries
#define WS_AT 64                     // A~ transposed: (P=100) x COLS=256
#define WS_PARTIAL (64 + NP * COLS)  // per-block partial sums

// ---------------------------------------------------------------------------
// Kernel 0: per-component prep.
//   Sigma_k = I + M_k^T M_k  -> Cholesky L -> logdet; A~_k = M_k L^{-T}
//   stored transposed (row p, col k*25+j) padded with zeros to 256 cols.
//   Also log_softmax(pi) and the Stiefel surface-area constant.
// ---------------------------------------------------------------------------
__global__ __launch_bounds__(128) void macg_prep(const float* __restrict__ M,
                                                 const float* __restrict__ pi,
                                                 float* __restrict__ ws) {
  const int k = blockIdx.x;
  const int tid = threadIdx.x;
  __shared__ float S[NR * NR];

  // Sigma = I + M^T M   (625 entries, 100-MAC dot each)
  const float* Mk = M + k * NP * NR;
  for (int idx = tid; idx < NR * NR; idx += blockDim.x) {
    int i = idx / NR, j = idx % NR;
    float acc = (i == j) ? 1.0f : 0.0f;
    for (int p = 0; p < NP; ++p) acc = fmaf(Mk[p * NR + i], Mk[p * NR + j], acc);
    S[idx] = acc;
  }
  __syncthreads();

  // Serial Cholesky (tiny: ~5K flops) in-place lower triangle; logdet.
  if (tid == 0) {
    float ldet = 0.0f;
    for (int j = 0; j < NR; ++j) {
      float s = S[j * NR + j];
      for (int r = 0; r < j; ++r) s -= S[j * NR + r] * S[j * NR + r];
      float ljj = sqrtf(s);
      S[j * NR + j] = ljj;
      ldet += 2.0f * logf(ljj);
      float inv = 1.0f / ljj;
      for (int i = j + 1; i < NR; ++i) {
        float t = S[i * NR + j];
        for (int r = 0; r < j; ++r) t -= S[i * NR + r] * S[j * NR + r];
        S[i * NR + j] = t * inv;
      }
    }
    ws[WS_LD + k] = ldet;
  }
  __syncthreads();

  // A~ rows: solve a L^T = m  (forward substitution per row, rows parallel)
  float* At = ws + WS_AT;
  if (tid < NP) {
    const int p = tid;
    const float* mrow = Mk + p * NR;
    float a[NR];
    for (int j = 0; j < NR; ++j) {
      float t = mrow[j];
      for (int r = 0; r < j; ++r) t -= a[r] * S[j * NR + r];
      a[j] = t / S[j * NR + j];
      At[p * COLS + k * NR + j] = a[j];
    }
  }

  if (k == 0) {
    // zero the 6 pad columns
    for (int idx = tid; idx < NP * (COLS - NK * NR); idx += blockDim.x) {
      int p = idx / (COLS - NK * NR), c = idx % (COLS - NK * NR);
      At[p * COLS + NK * NR + c] = 0.0f;
    }
    if (tid == 0) {
      // log_softmax(pi)
      float m = pi[0];
      for (int i = 1; i < NK; ++i) m = fmaxf(m, pi[i]);
      float se = 0.0f;
      for (int i = 0; i < NK; ++i) se += expf(pi[i] - m);
      float lse = m + logf(se);
      for (int i = 0; i < NK; ++i) ws[WS_LP + i] = pi[i] - lse;
      // logSA of Stiefel V_2(R^100): c=50, q=2
      const float LOG_PI = 1.1447298858494002f;
      const float LOG_2 = 0.6931471805599453f;
      float lgk = 0.5f * LOG_PI + lgammaf(50.0f) + lgammaf(49.5f);
      ws[WS_CONST0] = lgk - 2.0f * LOG_2 - 100.0f * LOG_PI;
    }
  }
}

// ---------------------------------------------------------------------------
// Kernel 1: main. One wave (one block) handles 16 samples as TWO 16-row
// A tiles (16 rows = 8 samples x q=2 each), sharing every B fragment:
//   D(32x256) = Xrows(32x100) * At(100x256), 50 WMMAs per 25 B loads.
// Epilogue per (sample,k): C = B~ B~^T (2x2) over that k's 25 columns,
// pdf = det(I-C) = 1 - tr(C) + det(C); logsumexp over k; per-block partial.
// ---------------------------------------------------------------------------
__global__ __launch_bounds__(32) void macg_main(const float* __restrict__ X,
                                                const float* __restrict__ ws,
                                                float* __restrict__ partial,
                                                int N) {
  const int lane = threadIdx.x & 31;
  const int n0 = blockIdx.x * SPW;  // first sample for this wave

  // LDS: first used to stage X (3200 floats), then reused as the D matrix
  // stored column-major (256 cols x 32 rows = 8192 floats = 32 KB).
  __shared__ float Dl[32 * COLS];
  __shared__ float vals[SPW * NK];
  __shared__ float res[SPW];

  const float* At = ws + WS_AT;
  const long long xlimit = (long long)N * (NP * NQ);

  // ---- stage 16 samples of X coalesced (3200 consecutive floats) ----
  // Stream-once data: non-temporal loads keep the shared At panel cache-hot.
  for (int j = 0; j < SPW * NP * NQ / 32; ++j) {
    int idx = lane + 32 * j;
    long long g = (long long)n0 * (NP * NQ) + idx;
    Dl[idx] = (g < xlimit) ? __builtin_nontemporal_load(&X[g]) : 0.0f;
  }
  __syncthreads();

  // ---- extract fp32 A-fragments per ISA 16x4 layout ----
  // lanes 0-15: rows M=0..15, K = {0,1}; lanes 16-31: same rows, K = {2,3}
  const int m = lane & 15;
  const int sA = m >> 1;       // sample within row-tile A (0..7)
  const int q = m & 1;         // column of X_n (q dimension)
  const int kk = (lane >> 4) * 2;
  v2f afragA[KSTEPS], afragB[KSTEPS];
#pragma unroll
  for (int kp = 0; kp < KSTEPS; ++kp) {
    int p = 4 * kp + kk;
    afragA[kp].x = Dl[sA * (NP * NQ) + p * NQ + q];
    afragA[kp].y = Dl[sA * (NP * NQ) + (p + 1) * NQ + q];
    afragB[kp].x = Dl[(8 + sA) * (NP * NQ) + p * NQ + q];
    afragB[kp].y = Dl[(8 + sA) * (NP * NQ) + (p + 1) * NQ + q];
  }
  __syncthreads();  // X staging consumed; LDS region becomes D buffer

  // ---- GEMM: 16 column tiles x 25 K-steps, each B fragment feeds 2 WMMAs --
  const int col_in_tile = lane & 15;
  const int h = lane >> 4;  // which half of D rows this lane covers
  for (int ct = 0; ct < 16; ++ct) {
    v8f accA = {};
    v8f accB = {};
#pragma unroll
    for (int kp = 0; kp < KSTEPS; ++kp) {
      int p0 = 4 * kp;
      v2f b;  // B 4x16 layout: VGPR0 = K row kk, VGPR1 = K row kk+1
      b.x = At[(p0 + kk) * COLS + ct * 16 + col_in_tile];
      b.y = At[(p0 + kk + 1) * COLS + ct * 16 + col_in_tile];
      accA = __builtin_amdgcn_wmma_f32_16x16x4_f32(
          false, afragA[kp], false, b, (short)0, accA, false, false);
      accB = __builtin_amdgcn_wmma_f32_16x16x4_f32(
          false, afragB[kp], false, b, (short)0, accB, false, false);
    }
    // Lane holds rows (v + 8h) of its column. Store column-major, tile A in
    // rows 0-15, tile B in rows 16-31: contiguous 4-float chunks -> b128.
    int col = ct * 16 + col_in_tile;
    v4f a_lo = {accA[0], accA[1], accA[2], accA[3]};
    v4f a_hi = {accA[4], accA[5], accA[6], accA[7]};
    v4f b_lo = {accB[0], accB[1], accB[2], accB[3]};
    v4f b_hi = {accB[4], accB[5], accB[6], accB[7]};
    *(v4f*)&Dl[col * 32 + 8 * h] = a_lo;
    *(v4f*)&Dl[col * 32 + 8 * h + 4] = a_hi;
    *(v4f*)&Dl[col * 32 + 16 + 8 * h] = b_lo;
    *(v4f*)&Dl[col * 32 + 16 + 8 * h + 4] = b_hi;
  }
  __syncthreads();

  // ---- epilogue: 160 (sample,k) tasks across 32 lanes ----
  const float c0 = ws[WS_CONST0];
  for (int i = 0; i < (SPW * NK + 31) / 32; ++i) {
    int idx = lane + 32 * i;
    if (idx < SPW * NK) {
      int ss = idx / NK, kc = idx % NK;
      float c00 = 0.f, c01 = 0.f, c11 = 0.f;
      int cbase = kc * NR;
      for (int c = 0; c < NR; ++c) {
        float d0 = Dl[(cbase + c) * 32 + 2 * ss];
        float d1 = Dl[(cbase + c) * 32 + 2 * ss + 1];
        c00 = fmaf(d0, d0, c00);
        c01 = fmaf(d0, d1, c01);
        c11 = fmaf(d1, d1, c11);
      }
      float pdf = 1.0f - (c00 + c11) + (c00 * c11 - c01 * c01);  // det(I-C)
      vals[idx] = c0 - ws[WS_LD + kc] - 50.0f * logf(pdf) + ws[WS_LP + kc];
    }
  }
  __syncthreads();

  // logsumexp over k per sample (16 lanes)
  if (lane < SPW) {
    float r = 0.0f;
    if (n0 + lane < N) {
      const float* v = &vals[lane * NK];
      float mx = v[0];
      for (int kc = 1; kc < NK; ++kc) mx = fmaxf(mx, v[kc]);
      float se = 0.0f;
      for (int kc = 0; kc < NK; ++kc) se += expf(v[kc] - mx);
      r = mx + logf(se);
    }
    res[lane] = r;
  }
  __syncthreads();

  if (lane == 0) {
    float t = 0.0f;
    for (int i = 0; i < SPW; ++i) t += res[i];
    partial[blockIdx.x] = t;  // fixed-order partial -> deterministic
  }
}

// ---------------------------------------------------------------------------
// Kernel 2: deterministic final reduction of per-block partials.
// ---------------------------------------------------------------------------
__global__ __launch_bounds__(256) void macg_reduce(const float* __restrict__ partial,
                                                   int nblocks,
                                                   float* __restrict__ out) {
  __shared__ float sm[256];
  int tid = threadIdx.x;
  float t = 0.0f;
  for (int i = tid; i < nblocks; i += 256) t += partial[i];
  sm[tid] = t;
  __syncthreads();
  for (int sft = 128; sft > 0; sft >>= 1) {
    if (tid < sft) sm[tid] += sm[tid + sft];
    __syncthreads();
  }
  if (tid == 0) out[0] = sm[0];
}

extern "C" void kernel_launch(void* const* d_in, const int* in_sizes, int n_in,
                              void* d_out, int out_size, void* d_ws, size_t ws_size,
                              hipStream_t stream) {
  const float* X = (const float*)d_in[0];   // (N, P, Q) fp32
  const float* M = (const float*)d_in[1];   // (K, P, R) fp32
  const float* pi = (const float*)d_in[2];  // (K,) fp32
  float* ws = (float*)d_ws;

  const int N = in_sizes[0] / (NP * NQ);
  const int nblocks = (N + SPW - 1) / SPW;  // 16 samples per single-wave block

  macg_prep<<<NK, 128, 0, stream>>>(M, pi, ws);
  macg_main<<<nblocks, 32, 0, stream>>>(X, ws, ws + WS_PARTIAL, N);
  macg_reduce<<<1, 256, 0, stream>>>(ws + WS_PARTIAL, nblocks, (float*)d_out);
}